// GraphAttention_90718299226204
// MI455X (gfx1250) — compile-verified
//
#include <hip/hip_runtime.h>
#include <hip/hip_bf16.h>
#include <math.h>

#define BSZ   8
#define NNODE 20000
#define F_IN  256
#define H     8
#define D     64
#define EEDGE 160000
#define NT16  (NNODE / 16)          // 1250 row-tiles per batch
#define WFRAG_PER_B (64 * 32 * 2)   // 64 k-steps * 32 lanes * float2 = 4096 floats

typedef __attribute__((ext_vector_type(2))) float v2f;
typedef __attribute__((ext_vector_type(8))) float v8f;

// ---------------------------------------------------------------------------
// Kernel A: w_eff[b,h,f] = sum_d scoring[b,h,d] * W_proj[h*64+d, f]
// Written directly in the WMMA B-fragment per-lane layout:
//   frag[b][step k4][lane].{x,y} = Wt[k4*4 + 2*(lane/16) + c][lane%16]
// where Wt[f][j]: j<8 -> src head j, j>=8 -> trg head j-8.
// grid = BSZ*H blocks, 256 threads (one per f).
// ---------------------------------------------------------------------------
__global__ void weff_swizzle_kernel(const float* __restrict__ Wp,
                                    const float* __restrict__ ssrc,
                                    const float* __restrict__ strg,
                                    float* __restrict__ wfrag) {
    const int f = threadIdx.x;              // 0..255
    const int b = blockIdx.x >> 3;
    const int h = blockIdx.x & 7;
    const float* sv = ssrc + (size_t)(b * H + h) * D;
    const float* tv = strg + (size_t)(b * H + h) * D;
    const float* wc = Wp + (size_t)(h * D) * F_IN + f;
    float as = 0.f, at = 0.f;
#pragma unroll 8
    for (int d = 0; d < D; ++d) {
        float w = wc[(size_t)d * F_IN];
        as += sv[d] * w;
        at += tv[d] * w;
    }
    // position of (f, j) inside the fragment table
    const int base = b * WFRAG_PER_B
                   + (((f >> 2) * 32 + ((f >> 1) & 1) * 16) << 1) + (f & 1);
    wfrag[base + (h << 1)]       = as;   // j = h      (src)
    wfrag[base + ((h + 8) << 1)] = at;   // j = h + 8  (trg)
}

// ---------------------------------------------------------------------------
// Kernel B: per-batch [20000 x 256] x [256 x 16] via V_WMMA_F32_16X16X4_F32.
// One wave per 16x16 tile; 64 k-steps. 4 waves / 128-thread block.
// Writes flat score tables ss[(b*N+n)*H+h], st[(b*N+n)*H+h].
// ---------------------------------------------------------------------------
__global__ __launch_bounds__(128)
void scores_wmma_kernel(const float* __restrict__ ch,
                        const float* __restrict__ wfrag,
                        float* __restrict__ ss,
                        float* __restrict__ st) {
    const int lane = threadIdx.x & 31;
    const int wave = threadIdx.x >> 5;
    const int tile = blockIdx.x * 4 + wave;          // 0 .. 9999 exactly
    const int b    = tile / NT16;
    const int m0   = (tile - b * NT16) * 16;

    // A fragment base: row m0 + lane%16, col 2*(lane/16)
    const float* aptr = ch
        + ((size_t)(b * NNODE + m0 + (lane & 15)) * F_IN) + ((lane >> 4) << 1);
    const v2f* bptr = (const v2f*)(wfrag + (size_t)b * WFRAG_PER_B) + lane;

    v8f acc = {};
#pragma unroll 8
    for (int s = 0; s < 64; ++s) {
        v2f a  = *(const v2f*)(aptr + 4 * s);        // k = 4*s (+ lane k-offset)
        v2f bb = bptr[s * 32];                        // coalesced b64, L2-resident
        acc = __builtin_amdgcn_wmma_f32_16x16x4_f32(
            /*neg_a=*/false, a, /*neg_b=*/false, bb,
            /*c_mod=*/(short)0, acc, /*reuse_a=*/false, /*reuse_b=*/false);
    }

    // D layout: VGPR v, lanes 0-15 -> row m0+v, lanes 16-31 -> row m0+v+8; col = lane%16
    const int j     = lane & 15;
    const int nbase = m0 + ((lane >> 4) << 3);
    float* dst      = (j < 8) ? ss : st;
    const size_t o  = ((size_t)b * NNODE + nbase) * H + (j & 7);
#pragma unroll
    for (int v = 0; v < 8; ++v)
        dst[o + (size_t)v * H] = acc[v];
}

// ---------------------------------------------------------------------------
// Kernel C: out[i] = sigmoid(ss[head[i]] + st[tail[i]])  (global flat gather)
// Score tables ~10 MB -> L2-resident on MI455X (192 MB L2).
// ---------------------------------------------------------------------------
__global__ void gather_sigmoid_kernel(const int* __restrict__ head,
                                      const int* __restrict__ tail,
                                      const float* __restrict__ ss,
                                      const float* __restrict__ st,
                                      float* __restrict__ out, int n) {
    int i = blockIdx.x * blockDim.x + threadIdx.x;
    if (i >= n) return;
    float x = ss[head[i]] + st[tail[i]];
    out[i] = 1.0f / (1.0f + __expf(-x));
}

extern "C" void kernel_launch(void* const* d_in, const int* in_sizes, int n_in,
                              void* d_out, int out_size, void* d_ws, size_t ws_size,
                              hipStream_t stream) {
    const float* ch   = (const float*)d_in[0];   // [BSZ, N, F_IN]
    const int*   head = (const int*)d_in[1];     // [BSZ, E]
    const int*   tail = (const int*)d_in[2];     // [BSZ, E]
    const float* Wp   = (const float*)d_in[3];   // [H*D, F_IN]
    const float* ssrc = (const float*)d_in[4];   // [BSZ, 1, H, D]
    const float* strg = (const float*)d_in[5];   // [BSZ, 1, H, D]
    float*       out  = (float*)d_out;           // [BSZ, E]

    // workspace layout
    float* wfrag = (float*)d_ws;                             // 8 * 4096 floats
    float* ss    = wfrag + (size_t)BSZ * WFRAG_PER_B;        // BSZ*N*H floats
    float* st    = ss + (size_t)BSZ * NNODE * H;             // BSZ*N*H floats

    weff_swizzle_kernel<<<BSZ * H, 256, 0, stream>>>(Wp, ssrc, strg, wfrag);

    // 10000 tiles total, 4 waves per 128-thread block
    scores_wmma_kernel<<<(BSZ * NT16) / 4, 128, 0, stream>>>(ch, wfrag, ss, st);

    const int n = BSZ * EEDGE;
    gather_sigmoid_kernel<<<(n + 255) / 256, 256, 0, stream>>>(head, tail, ss, st, out, n);
}